// SelfAttention_532575944749
// MI455X (gfx1250) — compile-verified
//
#include <hip/hip_runtime.h>
#include <math.h>

typedef __attribute__((ext_vector_type(2))) float v2f;
typedef __attribute__((ext_vector_type(8))) float v8f;

#define BATCH   32
#define DIM     4096
#define NHEADS  32
#define NKVH    8
#define HDIM    128
#define TTOT    2048
#define NEWT    2047      // start_pos (compile-time constant per reference)
#define CHUNK   128       // KV rows staged in LDS per iteration
#define KROWW   132       // padded row stride in floats (528 B, 16B-aligned)

// ---------------------------------------------------------------------------
// CDNA5 async-copy helpers (inline asm; portable across ROCm7.2 / clang-23).
// One instruction moves a full 512B KV row: lane l copies 16 bytes.
// LDS address is taken from the VDST VGPR; low 32 bits of a generic LDS pointer are the
// LDS byte offset (ISA 10.2 aperture rules).
// ---------------------------------------------------------------------------
__device__ __forceinline__ void async_row_to_lds(const float* __restrict__ gsrc,
                                                 float* lds_row)
{
    const int lane = threadIdx.x & 31;
    unsigned long long ga = (unsigned long long)(uintptr_t)gsrc + (unsigned)(lane * 16);
    unsigned la = (unsigned)(uintptr_t)lds_row + (unsigned)(lane * 16);
    asm volatile("global_load_async_to_lds_b128 %0, %1, off"
                 :: "v"(la), "v"(ga) : "memory");
}

__device__ __forceinline__ void wait_async0()
{
    asm volatile("s_wait_asynccnt 0" ::: "memory");
}

// ---------------------------------------------------------------------------
// Skinny GEMM: Out[m,n] = sum_k X[m,k] * W[n,k],  M == 32 (batch), W row-major
// over k.  LDS-staged, computed with V_WMMA_F32_16X16X4_F32.
// Block: 256 threads = 8 waves; each wave owns a 16-wide column strip,
// two 16-row M tiles (rows 0-15 / 16-31).  BN = 128 cols/block, BK = 32.
// ---------------------------------------------------------------------------
__global__ __launch_bounds__(256)
void gemm32_wmma_f32(const float* __restrict__ X, const float* __restrict__ W,
                     float* __restrict__ Out, int K, int N)
{
    __shared__ float w_lds[128][33];   // padded: stride 33 -> conflict-free frag reads
    __shared__ float x_lds[32][33];

    const int tid  = threadIdx.x;
    const int wv   = tid >> 5;         // wave id 0..7
    const int lane = tid & 31;
    const int hv   = lane >> 4;        // lane half: selects K pair {0,1} vs {2,3}
    const int l    = lane & 15;
    const int n0   = blockIdx.x * 128;

    v8f acc0 = {};                      // M rows 0..15
    v8f acc1 = {};                      // M rows 16..31

    const int r8 = tid >> 3;            // 0..31
    const int c4 = (tid & 7) * 4;       // 0..28

    for (int k0 = 0; k0 < K; k0 += 32) {
        // Cooperative coalesced load of W tile (128 x 32) into LDS.
#pragma unroll
        for (int i = 0; i < 4; ++i) {
            const int row = r8 + 32 * i;
            const float* wp = W + (size_t)(n0 + row) * K + k0 + c4;
            const float4 w4 = *(const float4*)wp;
            if (k0 + 32 < K) __builtin_prefetch(wp + 32, 0, 0);  // next k-tile
            w_lds[row][c4 + 0] = w4.x; w_lds[row][c4 + 1] = w4.y;
            w_lds[row][c4 + 2] = w4.z; w_lds[row][c4 + 3] = w4.w;
        }
        // X tile (32 x 32): exactly one float4 per thread.
        {
            const float4 x4 = *(const float4*)(X + (size_t)r8 * K + k0 + c4);
            x_lds[r8][c4 + 0] = x4.x; x_lds[r8][c4 + 1] = x4.y;
            x_lds[r8][c4 + 2] = x4.z; x_lds[r8][c4 + 3] = x4.w;
        }
        __syncthreads();

#pragma unroll
        for (int kk = 0; kk < 32; kk += 4) {
            const int kb = kk + 2 * hv;
            v2f a0, a1, bfr;
            // A 16x4 fragment (ISA 7.12.2): lanes 0-15 K={0,1}, lanes 16-31 K={2,3}
            a0[0]  = x_lds[l][kb];          a0[1]  = x_lds[l][kb + 1];
            a1[0]  = x_lds[16 + l][kb];     a1[1]  = x_lds[16 + l][kb + 1];
            // B 4x16 fragment: n = lane%16, lane half selects K pair; B[k][n] = W[n][k]
            bfr[0] = w_lds[wv * 16 + l][kb]; bfr[1] = w_lds[wv * 16 + l][kb + 1];

            acc0 = __builtin_amdgcn_wmma_f32_16x16x4_f32(
                false, a0, false, bfr, (short)0, acc0, false, false);
            acc1 = __builtin_amdgcn_wmma_f32_16x16x4_f32(
                false, a1, false, bfr, (short)0, acc1, false, false);
        }
        __syncthreads();
    }

    // D layout: VGPR j -> (M=j, N=l) lanes 0-15; (M=j+8, N=l) lanes 16-31.
    const int n = n0 + wv * 16 + l;
#pragma unroll
    for (int j = 0; j < 8; ++j) {
        const int m = j + 8 * hv;
        Out[(size_t)m * N + n]        = acc0[j];
        Out[(size_t)(m + 16) * N + n] = acc1[j];
    }
}

// ---------------------------------------------------------------------------
// RoPE on xq (32x32x128) and xk (32x8x128), in workspace, in place.
// ---------------------------------------------------------------------------
__global__ void rope_kernel(float* __restrict__ xq, float* __restrict__ xk,
                            const float* __restrict__ fcos,
                            const float* __restrict__ fsin)
{
    const int NQ = BATCH * NHEADS * (HDIM / 2);
    const int NK = BATCH * NKVH   * (HDIM / 2);
    int j = blockIdx.x * blockDim.x + threadIdx.x;
    float* buf;
    if (j < NQ)            { buf = xq; }
    else if (j < NQ + NK)  { buf = xk; j -= NQ; }
    else return;
    const int p = j & 63;
    const float c = fcos[p], s = fsin[p];
    const float x0 = buf[2 * j], x1 = buf[2 * j + 1];
    buf[2 * j]     = x0 * c - x1 * s;
    buf[2 * j + 1] = x0 * s + x1 * c;
}

// ---------------------------------------------------------------------------
// GQA decode attention: one block per (b, kv-head g).
// K/V rows are streamed into LDS with GLOBAL_LOAD_ASYNC_TO_LDS_B128 in 128-row
// chunks (one async instruction per 512B row, ASYNCcnt-tracked), then consumed
// from LDS.  Two-pass softmax over T=2048 with scores resident in LDS.
// t == 2047 sources the freshly computed (roped) k / v from workspace so the
// input caches are never mutated.
// ---------------------------------------------------------------------------
__global__ __launch_bounds__(256)
void attn_kernel(const float* __restrict__ xq, const float* __restrict__ xk,
                 const float* __restrict__ xv, const float* __restrict__ cache_k,
                 const float* __restrict__ cache_v, float* __restrict__ attn_out)
{
    __shared__ float q_s[4 * HDIM];          //   2 KB
    __shared__ float sc[4 * TTOT];           //  32 KB
    __shared__ float red[4 * 256];           //   4 KB
    __shared__ float part[8][4 * HDIM];      //  16 KB
    __shared__ float kbuf[CHUNK * KROWW];    //  66 KB async staging buffer
    __shared__ float mx_s[4], inv_s[4];

    const int tid  = threadIdx.x;
    const int w    = tid >> 5;
    const int lane = tid & 31;
    const int b    = blockIdx.x >> 3;
    const int g    = blockIdx.x & 7;
    const float scale = 0.08838834764831845f;  // 1/sqrt(128)

    const float* knew = xk + ((size_t)b * NKVH + g) * HDIM;
    const float* vnew = xv + ((size_t)b * NKVH + g) * HDIM;

    // Load the 4 query heads of this group, pre-scaled.
    for (int e = tid; e < 4 * HDIM; e += 256) {
        const int r = e >> 7, d = e & 127;
        q_s[e] = xq[((size_t)b * NHEADS + (g * 4 + r)) * HDIM + d] * scale;
    }
    __syncthreads();

    // ---- Pass 1: scores[r][t] = q_r . k_t, K rows async-staged in LDS ----
    for (int ch = 0; ch < TTOT / CHUNK; ++ch) {
        // Stage: wave w async-copies rows w*16 .. w*16+15 of this chunk.
#pragma unroll
        for (int i = 0; i < 16; ++i) {
            const int row = w * 16 + i;
            const int t   = ch * CHUNK + row;
            const float* src = (t == NEWT)
                ? knew : (cache_k + (((size_t)b * TTOT + t) * NKVH + g) * HDIM);
            async_row_to_lds(src, &kbuf[row * KROWW]);
        }
        wait_async0();
        __syncthreads();

        // Compute: two threads per row (halves of d); combine with one shuffle.
        {
            const int row = tid >> 1;
            const int hs  = tid & 1;             // which 64-float half of the row
            const int t   = ch * CHUNK + row;
            const float* kr = &kbuf[row * KROWW + hs * 64];
            const float* qb = q_s + hs * 64;
            float s0 = 0.f, s1 = 0.f, s2 = 0.f, s3 = 0.f;
#pragma unroll
            for (int c = 0; c < 64; c += 4) {
                const float4 k4 = *(const float4*)(kr + c);
                const float4 q0 = *(const float4*)(qb + 0 * HDIM + c);
                const float4 q1 = *(const float4*)(qb + 1 * HDIM + c);
                const float4 q2 = *(const float4*)(qb + 2 * HDIM + c);
                const float4 q3 = *(const float4*)(qb + 3 * HDIM + c);
                s0 += q0.x * k4.x + q0.y * k4.y + q0.z * k4.z + q0.w * k4.w;
                s1 += q1.x * k4.x + q1.y * k4.y + q1.z * k4.z + q1.w * k4.w;
                s2 += q2.x * k4.x + q2.y * k4.y + q2.z * k4.z + q2.w * k4.w;
                s3 += q3.x * k4.x + q3.y * k4.y + q3.z * k4.z + q3.w * k4.w;
            }
            s0 += __shfl_xor(s0, 1, 32);
            s1 += __shfl_xor(s1, 1, 32);
            s2 += __shfl_xor(s2, 1, 32);
            s3 += __shfl_xor(s3, 1, 32);
            if (hs == 0) {
                sc[0 * TTOT + t] = s0; sc[1 * TTOT + t] = s1;
                sc[2 * TTOT + t] = s2; sc[3 * TTOT + t] = s3;
            }
        }
        __syncthreads();   // kbuf reads done before next chunk overwrites it
    }

    // ---- Row max ----
    {
        float m[4] = {-1e30f, -1e30f, -1e30f, -1e30f};
        for (int t = tid; t < TTOT; t += 256)
#pragma unroll
            for (int r = 0; r < 4; ++r) m[r] = fmaxf(m[r], sc[r * TTOT + t]);
#pragma unroll
        for (int r = 0; r < 4; ++r) red[r * 256 + tid] = m[r];
        __syncthreads();
        for (int s = 128; s > 0; s >>= 1) {
            if (tid < s)
#pragma unroll
                for (int r = 0; r < 4; ++r)
                    red[r * 256 + tid] = fmaxf(red[r * 256 + tid], red[r * 256 + tid + s]);
            __syncthreads();
        }
        if (tid < 4) mx_s[tid] = red[tid * 256];
        __syncthreads();
    }

    // ---- exp + row sum (exp written back into sc) ----
    {
        float sm[4] = {0.f, 0.f, 0.f, 0.f};
        for (int t = tid; t < TTOT; t += 256)
#pragma unroll
            for (int r = 0; r < 4; ++r) {
                const float e = __expf(sc[r * TTOT + t] - mx_s[r]);
                sc[r * TTOT + t] = e;
                sm[r] += e;
            }
        __syncthreads();   // red[] reuse
#pragma unroll
        for (int r = 0; r < 4; ++r) red[r * 256 + tid] = sm[r];
        __syncthreads();
        for (int s = 128; s > 0; s >>= 1) {
            if (tid < s)
#pragma unroll
                for (int r = 0; r < 4; ++r)
                    red[r * 256 + tid] += red[r * 256 + tid + s];
            __syncthreads();
        }
        if (tid < 4) inv_s[tid] = 1.0f / red[tid * 256];
        __syncthreads();
    }

    // ---- Pass 2: out[r][:] = sum_t p[r][t] * v_t, V rows async-staged ----
    {
        float4 acc[4] = {};
        for (int ch = 0; ch < TTOT / CHUNK; ++ch) {
#pragma unroll
            for (int i = 0; i < 16; ++i) {
                const int row = w * 16 + i;
                const int t   = ch * CHUNK + row;
                const float* src = (t == NEWT)
                    ? vnew : (cache_v + (((size_t)b * TTOT + t) * NKVH + g) * HDIM);
                async_row_to_lds(src, &kbuf[row * KROWW]);
            }
            wait_async0();
            __syncthreads();

            // Wave w consumes rows w*16..w*16+15; lane covers d = lane*4..+3.
#pragma unroll
            for (int i = 0; i < 16; ++i) {
                const int row = w * 16 + i;
                const int t   = ch * CHUNK + row;
                const float4 v4 = *(const float4*)(&kbuf[row * KROWW + lane * 4]);
#pragma unroll
                for (int r = 0; r < 4; ++r) {
                    const float p = sc[r * TTOT + t] * inv_s[r];
                    acc[r].x += p * v4.x; acc[r].y += p * v4.y;
                    acc[r].z += p * v4.z; acc[r].w += p * v4.w;
                }
            }
            __syncthreads();
        }
#pragma unroll
        for (int r = 0; r < 4; ++r)
            *(float4*)&part[w][r * HDIM + lane * 4] = acc[r];
        __syncthreads();
        for (int e = tid; e < 4 * HDIM; e += 256) {
            float sum = 0.f;
#pragma unroll
            for (int w2 = 0; w2 < 8; ++w2) sum += part[w2][e];
            const int r = e >> 7, d = e & 127;
            attn_out[((size_t)b * NHEADS + (g * 4 + r)) * HDIM + d] = sum;
        }
    }
}

// ---------------------------------------------------------------------------
extern "C" void kernel_launch(void* const* d_in, const int* in_sizes, int n_in,
                              void* d_out, int out_size, void* d_ws, size_t ws_size,
                              hipStream_t stream)
{
    (void)in_sizes; (void)n_in; (void)out_size; (void)ws_size;
    const float* x    = (const float*)d_in[0];
    const float* fcos = (const float*)d_in[1];
    const float* fsin = (const float*)d_in[2];
    const float* ck   = (const float*)d_in[3];
    const float* cv   = (const float*)d_in[4];
    const float* wq   = (const float*)d_in[5];
    const float* wk   = (const float*)d_in[6];
    const float* wvp  = (const float*)d_in[7];
    const float* wo   = (const float*)d_in[8];
    float* out = (float*)d_out;

    float* xq = (float*)d_ws;                       // 32*4096
    float* xk = xq + BATCH * NHEADS * HDIM;         // 32*1024
    float* xv = xk + BATCH * NKVH * HDIM;           // 32*1024
    float* ao = xv + BATCH * NKVH * HDIM;           // 32*4096

    // QKV projections (WMMA f32)
    gemm32_wmma_f32<<<DIM / 128, 256, 0, stream>>>(x, wq, xq, DIM, NHEADS * HDIM);
    gemm32_wmma_f32<<<(NKVH * HDIM) / 128, 256, 0, stream>>>(x, wk, xk, DIM, NKVH * HDIM);
    gemm32_wmma_f32<<<(NKVH * HDIM) / 128, 256, 0, stream>>>(x, wvp, xv, DIM, NKVH * HDIM);

    // RoPE on q and k
    const int npairs = BATCH * NHEADS * (HDIM / 2) + BATCH * NKVH * (HDIM / 2);
    rope_kernel<<<(npairs + 255) / 256, 256, 0, stream>>>(xq, xk, fcos, fsin);

    // Attention: one block per (batch, kv-head), async KV staging
    attn_kernel<<<BATCH * NKVH, 256, 0, stream>>>(xq, xk, xv, ck, cv, ao);

    // Output projection (WMMA f32)
    gemm32_wmma_f32<<<DIM / 128, 256, 0, stream>>>(ao, wo, out, DIM, DIM);
}